// Emu3p5VisionVQVectorQuantizer_88115549045209
// MI455X (gfx1250) — compile-verified
//
#include <hip/hip_runtime.h>

typedef __attribute__((ext_vector_type(2))) float v2f;
typedef __attribute__((ext_vector_type(8))) float v8f;

#define NE      32768   // codebook entries
#define DDIM    32      // embedding dim (= K)
#define MTOT    4096    // B*H*W spatial positions
#define HW      1024    // H*W
#define WGS     512     // threads per block
#define NWAVES  16      // waves per block
#define MBLK    32      // spatial positions per block (2 WMMA M-tiles)
#define NCHUNK  (NE / NWAVES)   // 2048 codebook rows per wave

__global__ __launch_bounds__(WGS)
void vq_wmma_argmax_kernel(const float* __restrict__ z,
                           const float* __restrict__ emb,
                           float* __restrict__ zq_out,
                           float* __restrict__ ind_out)
{
    __shared__ float s_val[NWAVES][MBLK];
    __shared__ int   s_idx[NWAVES][MBLK];
    __shared__ int   s_win[MBLK];

    const int tid  = threadIdx.x;
    const int lane = tid & 31;
    const int wave = tid >> 5;      // 0..15
    const int lh   = lane & 15;     // lane within half
    const int hi   = lane >> 4;     // which half (K-offset selector)

    const int m0 = blockIdx.x * MBLK;      // first spatial position of this block
    const int bb = m0 >> 10;               // batch index (blocks never cross batches)
    const int sb = m0 & (HW - 1);          // h*W+w base

    // ---- Load A fragments for BOTH M-tiles once, reused for all N ----
    // V_WMMA_F32_16X16X4_F32 A layout: lanes 0-15 hold {K=k0,k0+1}, lanes 16-31 {k0+2,k0+3}
    const float* zp0 = z + (size_t)bb * (DDIM * HW) + sb + lh;        // tile0 row
    const float* zp1 = zp0 + 16;                                      // tile1 row
    v2f a0[8], a1[8];
#pragma unroll
    for (int j = 0; j < 8; ++j) {
        const int k = 4 * j + 2 * hi;
        a0[j].x = zp0[(size_t)k * HW];
        a0[j].y = zp0[(size_t)(k + 1) * HW];
        a1[j].x = zp1[(size_t)k * HW];
        a1[j].y = zp1[(size_t)(k + 1) * HW];
    }

    // Running per-lane argmax: cT[r] = logits[m0 + 16*T + r + 8*hi][n + lh]
    float bval0[8], bval1[8];
    int   bidx0[8], bidx1[8];
#pragma unroll
    for (int r = 0; r < 8; ++r) {
        bval0[r] = -__builtin_inff(); bidx0[r] = 0;
        bval1[r] = -__builtin_inff(); bidx1[r] = 0;
    }

    const int n_begin = wave * NCHUNK;
    const int n_end   = n_begin + NCHUNK;

    // One B tile per iteration feeds TWO independent accumulator chains
    // (16 WMMAs : 8 loads) -> half the L2 traffic and VMEM issue per WMMA.
    for (int n0 = n_begin; n0 < n_end; n0 += 16) {
        const float* erow = emb + (size_t)(n0 + lh) * DDIM + 2 * hi;

        v2f bf[8];
#pragma unroll
        for (int j = 0; j < 8; ++j) bf[j] = *(const v2f*)(erow + 4 * j);

        v8f c0 = {};
        v8f c1 = {};
#pragma unroll
        for (int j = 0; j < 8; ++j) {
            c0 = __builtin_amdgcn_wmma_f32_16x16x4_f32(
                     false, a0[j], false, bf[j], (short)0, c0, false, false);
            c1 = __builtin_amdgcn_wmma_f32_16x16x4_f32(
                     false, a1[j], false, bf[j], (short)0, c1, false, false);
        }

        const int n = n0 + lh;
#pragma unroll
        for (int r = 0; r < 8; ++r) {
            if (c0[r] > bval0[r]) { bval0[r] = c0[r]; bidx0[r] = n; }
        }
#pragma unroll
        for (int r = 0; r < 8; ++r) {
            if (c1[r] > bval1[r]) { bval1[r] = c1[r]; bidx1[r] = n; }
        }
    }

    // ---- Argmax reduction across the 16 lanes of each half (ties -> lower index) ----
#pragma unroll
    for (int off = 8; off >= 1; off >>= 1) {
#pragma unroll
        for (int r = 0; r < 8; ++r) {
            float ov0 = __shfl_xor(bval0[r], off, 32);
            int   oi0 = __shfl_xor(bidx0[r], off, 32);
            if (ov0 > bval0[r] || (ov0 == bval0[r] && oi0 < bidx0[r])) {
                bval0[r] = ov0; bidx0[r] = oi0;
            }
            float ov1 = __shfl_xor(bval1[r], off, 32);
            int   oi1 = __shfl_xor(bidx1[r], off, 32);
            if (ov1 > bval1[r] || (ov1 == bval1[r] && oi1 < bidx1[r])) {
                bval1[r] = ov1; bidx1[r] = oi1;
            }
        }
    }

    if (lh == 0) {
#pragma unroll
        for (int r = 0; r < 8; ++r) {
            s_val[wave][r + 8 * hi]      = bval0[r];
            s_idx[wave][r + 8 * hi]      = bidx0[r];
            s_val[wave][16 + r + 8 * hi] = bval1[r];
            s_idx[wave][16 + r + 8 * hi] = bidx1[r];
        }
    }
    __syncthreads();

    // ---- Cross-wave reduction (16 candidates per row), write indices ----
    if (tid < MBLK) {
        float bv = s_val[0][tid];
        int   bi = s_idx[0][tid];
#pragma unroll
        for (int w = 1; w < NWAVES; ++w) {
            float v = s_val[w][tid];
            int   i = s_idx[w][tid];
            if (v > bv || (v == bv && i < bi)) { bv = v; bi = i; }
        }
        s_win[tid] = bi;
        ind_out[m0 + tid] = (float)bi;   // indices exact in fp32 (< 32768)
    }
    __syncthreads();

    // ---- Gather winning codebook rows into [B, D, H, W] (coalesced along w) ----
    for (int e = tid; e < MBLK * DDIM; e += WGS) {
        const int j = e & (MBLK - 1);   // position within block
        const int d = e >> 5;           // channel
        zq_out[(size_t)bb * (DDIM * HW) + (size_t)d * HW + sb + j] =
            emb[(size_t)s_win[j] * DDIM + d];
    }
}

extern "C" void kernel_launch(void* const* d_in, const int* in_sizes, int n_in,
                              void* d_out, int out_size, void* d_ws, size_t ws_size,
                              hipStream_t stream)
{
    const float* z   = (const float*)d_in[0];   // [4, 32, 32, 32] fp32
    const float* emb = (const float*)d_in[1];   // [32768, 32] fp32
    float* out = (float*)d_out;                 // 131072 z_q floats + 4096 indices

    vq_wmma_argmax_kernel<<<MTOT / MBLK, WGS, 0, stream>>>(z, emb, out, out + MTOT * DDIM);
}